// RPN_36558761624221
// MI455X (gfx1250) — compile-verified
//
#include <hip/hip_runtime.h>
#include <cstdint>
#include <cstddef>

typedef unsigned int        u32;
typedef unsigned long long  u64;
typedef unsigned short      u16;

typedef float    v2f __attribute__((ext_vector_type(2)));
typedef float    v8f __attribute__((ext_vector_type(8)));
typedef unsigned v4u __attribute__((ext_vector_type(4)));
typedef int      v4i __attribute__((ext_vector_type(4)));
typedef int      v8i __attribute__((ext_vector_type(8)));

#define NIMG   8
#define KSEL   4768
#define POST   1000
#define NMSTH  0.7f
#define BBOXCLIP 4.135166556742356f

__constant__ int c_feat[5]   = {256,128,64,32,16};
__constant__ int c_npl[5]    = {196608,49152,12288,3072,768};
__constant__ int c_lvloff[5] = {0,196608,245760,258048,261120};
__constant__ int c_seloff[5] = {0,1000,2000,3000,4000};
__constant__ int c_sel[5]    = {1000,1000,1000,1000,768};

#if defined(__has_builtin)
#if __has_builtin(__builtin_amdgcn_wmma_f32_16x16x4_f32)
#define HAS_WMMA4 1
#endif
#endif

__device__ __forceinline__ u32 fkey(float v) {
    u32 b = __float_as_uint(v);
    return (b & 0x80000000u) ? ~b : (b | 0x80000000u);
}

// ---------------------------------------------------------------------------
// 1) 4-pass byte radix SELECT per (image, level 0..3): find u32 key threshold
//    T such that count(u > T) = G, and neq = 1000 - G taken from u == T.
// ---------------------------------------------------------------------------
__global__ __launch_bounds__(256) void radix_kernel(const float* o0, const float* o1,
                                                    const float* o2, const float* o3,
                                                    u32* thr, u32* neq, u32* gcnt) {
    int n = blockIdx.x, l = blockIdx.y;           // l in 0..3
    const float* objs[4] = {o0, o1, o2, o3};
    int f  = c_feat[l];
    int nl = 3 * f * f;
    const float* src = objs[l] + (size_t)n * nl;
    int tid = threadIdx.x;

    __shared__ u32 hist[256];
    __shared__ u32 s_sel, s_above;

    u32 prefix = 0; int krem = 1000; u32 G = 0;
    for (int p = 3; p >= 0; --p) {
        for (int b = tid; b < 256; b += 256) hist[b] = 0;
        __syncthreads();
        for (int e = tid; e < nl; e += 256) {
            __builtin_prefetch(src + e + 4096, 0, 1);      // global_prefetch_b8
            u32 u = fkey(src[e]);
            bool ok = (p == 3) || ((u >> (8 * (p + 1))) == prefix);
            if (ok) atomicAdd(&hist[(u >> (8 * p)) & 0xFFu], 1u);
        }
        __syncthreads();
        if (tid == 0) {
            u32 above = 0; u32 sel = 0;
            for (int b = 255; b >= 0; --b) {
                u32 c = hist[b];
                if (above + c >= (u32)krem) { sel = (u32)b; break; }
                above += c;
            }
            s_sel = sel; s_above = above;
        }
        __syncthreads();
        prefix = (prefix << 8) | s_sel;
        G += s_above;
        krem -= (int)s_above;
        __syncthreads();
    }
    if (tid == 0) {
        thr [n * 5 + l] = prefix;
        neq [n * 5 + l] = (u32)krem;
        gcnt[n * 5 + l] = G;
    }
}

// ---------------------------------------------------------------------------
// 2) Collect selected anchor indices per (image, level) into candidate slots.
//    u > T via atomics (order fixed later by sort); u == T: smallest indices.
// ---------------------------------------------------------------------------
#define EQCAP 4096
__global__ __launch_bounds__(256) void collect_kernel(const float* o0, const float* o1,
                                                      const float* o2, const float* o3,
                                                      const u32* thr, const u32* neq,
                                                      const u32* gcnt, u32* cand) {
    int n = blockIdx.x, l = blockIdx.y;
    int tid = threadIdx.x;
    if (l == 4) {                                   // 768 <= 1000: take everything
        for (int t = tid; t < 768; t += 256)
            cand[(size_t)n * KSEL + c_seloff[4] + t] = (u32)(c_lvloff[4] + t);
        return;
    }
    const float* objs[4] = {o0, o1, o2, o3};
    int f  = c_feat[l];
    int nl = 3 * f * f;
    const float* src = objs[l] + (size_t)n * nl;
    u32 T = thr[n * 5 + l], NE = neq[n * 5 + l], G = gcnt[n * 5 + l];

    __shared__ u32 cgt, ceq;
    __shared__ u32 eq[EQCAP];
    if (tid == 0) { cgt = 0; ceq = 0; }
    __syncthreads();

    u32* out = cand + (size_t)n * KSEL + c_seloff[l];
    for (int e = tid; e < nl; e += 256) {
        u32 u = fkey(src[e]);
        if (u > T) {
            u32 pos = atomicAdd(&cgt, 1u);
            out[pos] = (u32)(c_lvloff[l] + e);
        } else if (u == T) {
            u32 p = atomicAdd(&ceq, 1u);
            if (p < EQCAP) eq[p] = (u32)e;
        }
    }
    __syncthreads();
    u32 ecnt = ceq < EQCAP ? ceq : EQCAP;
    for (u32 i = tid; i < EQCAP; i += 256) if (i >= ecnt) eq[i] = 0xFFFFFFFFu;
    __syncthreads();
    // bitonic ascending over eq[EQCAP] -> smallest indices first (top_k tiebreak)
    for (u32 sz = 2; sz <= EQCAP; sz <<= 1) {
        for (u32 st = sz >> 1; st > 0; st >>= 1) {
            for (u32 i = tid; i < EQCAP; i += 256) {
                u32 ixj = i ^ st;
                if (ixj > i) {
                    bool up = ((i & sz) == 0);
                    u32 a = eq[i], b = eq[ixj];
                    if (up ? (a > b) : (a < b)) { eq[i] = b; eq[ixj] = a; }
                }
            }
            __syncthreads();
        }
    }
    for (u32 c = tid; c < NE; c += 256)
        out[G + c] = (u32)c_lvloff[l] + eq[c];
}

// ---------------------------------------------------------------------------
// 3) Decode: sigmoid score, torchvision box decode, clip, validity, sort key.
// ---------------------------------------------------------------------------
__global__ __launch_bounds__(256) void decode_kernel(const float* o0, const float* o1, const float* o2,
                                                     const float* o3, const float* o4,
                                                     const float* d0, const float* d1, const float* d2,
                                                     const float* d3, const float* d4,
                                                     const float* anchors, const u32* cand,
                                                     float* boxesG, u64* key64, u32* keepOut) {
    int gid = blockIdx.x * 256 + threadIdx.x;
    if (gid >= NIMG * KSEL) return;
    keepOut[gid] = 0u;
    int n = gid / KSEL, slot = gid % KSEL;
    int l = slot < 1000 ? 0 : slot < 2000 ? 1 : slot < 3000 ? 2 : slot < 4000 ? 3 : 4;
    const float* objs[5] = {o0, o1, o2, o3, o4};
    const float* dels[5] = {d0, d1, d2, d3, d4};

    u32 gidx = cand[gid];
    int f  = c_feat[l];
    int t  = (int)gidx - c_lvloff[l];
    int a  = t % 3, hw = t / 3;
    int h  = hw / f, w = hw % f;

    float logit = objs[l][(((size_t)n * 3 + a) * f + h) * (size_t)f + w];
    float score = 1.0f / (1.0f + expf(-logit));

    size_t chs  = (size_t)f * f;
    size_t base = (((size_t)n * 12 + a * 4) * f + h) * (size_t)f + w;
    const float* dp = dels[l];
    float dx = dp[base], dy = dp[base + chs];
    float dw = fminf(dp[base + 2 * chs], BBOXCLIP);
    float dh = fminf(dp[base + 3 * chs], BBOXCLIP);

    const float* an = anchors + (size_t)gidx * 4;
    float wa = an[2] - an[0], ha = an[3] - an[1];
    float cx = an[0] + 0.5f * wa, cy = an[1] + 0.5f * ha;
    float pcx = dx * wa + cx, pcy = dy * ha + cy;
    float pw = expf(dw) * wa, ph = expf(dh) * ha;
    float x1 = pcx - 0.5f * pw, y1 = pcy - 0.5f * ph;
    float x2 = pcx + 0.5f * pw, y2 = pcy + 0.5f * ph;
    x1 = fminf(fmaxf(x1, 0.f), 1024.f); y1 = fminf(fmaxf(y1, 0.f), 1024.f);
    x2 = fminf(fmaxf(x2, 0.f), 1024.f); y2 = fminf(fmaxf(y2, 0.f), 1024.f);
    bool valid = (x2 - x1 >= 1e-3f) && (y2 - y1 >= 1e-3f) && (score >= 0.f);

    float* bo = boxesG + (size_t)gid * 4;
    bo[0] = x1; bo[1] = y1; bo[2] = x2; bo[3] = y2;

    u32 su = valid ? (__float_as_uint(score) | 0x80000000u) : 0u;
    u32 invg = (0x3FFFFu - gidx) & 0x3FFFFu;         // smaller gidx wins ties
    key64[gid] = ((u64)su << 32) | ((u64)invg << 13) | (u32)slot;
}

// ---------------------------------------------------------------------------
// 4) Per-image sort: TDM-stage the key array into LDS, bitonic sort desc,
//    emit sorted slot order + per-level candidate lists.
//    TDM path is UNCONDITIONAL in the device pass this round: compile
//    success proves tensor_load_to_lds lowered for gfx1250.
// ---------------------------------------------------------------------------
#define NPAD 8192
__global__ __launch_bounds__(1024) void sort_kernel(const u64* key64G, u32* sortedSlot,
                                                    u32* lvlList, u32* lvlCnt) {
    __shared__ u64 skeys[NPAD];                      // 64KB
    int n = blockIdx.x, tid = threadIdx.x;
    const u64* src = key64G + (size_t)n * KSEL;

#if defined(__AMDGCN__)
    if (tid == 0) {
        u32 lds_off = (u32)(u64)(void*)&skeys[0];
        u64 ga = (u64)(const void*)src;
        v4u g0 = { 1u,                                    // count=1, user mode
                   lds_off,
                   (u32)ga,
                   (u32)((ga >> 32) & 0x01FFFFFFu) | 0x80000000u };   // type=2
        // 1-row tile of 4768 8-byte elements (data_size code 3)
        v8i g1 = { (int)(3u << 16),                       // mask=0, data_size=8B
                   (int)(((u32)KSEL & 0xFFFFu) << 16),    // tensor_dim0 lo16
                   (int)(1u << 16),                       // tensor_dim0 hi16=0, tensor_dim1=1 lo16
                   (int)((u32)KSEL << 16),                // tensor_dim1 hi16=0, tile_dim0
                   0,                                     // tile_dim1=0, tile_dim2=0
                   (int)(u32)KSEL,                        // tensor_dim0_stride lo32
                   0, 0 };
        v4i gz = {0, 0, 0, 0};
#if __clang_major__ >= 23
        v8i gz8 = {0, 0, 0, 0, 0, 0, 0, 0};
        __builtin_amdgcn_tensor_load_to_lds(g0, g1, gz, gz, gz8, 0);
#else
        __builtin_amdgcn_tensor_load_to_lds(g0, g1, gz, gz, 0);
#endif
        __builtin_amdgcn_s_wait_tensorcnt(0);
    }
    for (int i = tid; i < NPAD - KSEL; i += 1024) skeys[KSEL + i] = 0ull;
    __syncthreads();
#else
    // host-pass placeholder only (never codegen'd for gfx1250)
    for (int i = tid; i < NPAD; i += 1024) skeys[i] = (i < KSEL) ? src[i] : 0ull;
    __syncthreads();
#endif

    // bitonic sort, descending
    for (u32 sz = 2; sz <= NPAD; sz <<= 1) {
        for (u32 st = sz >> 1; st > 0; st >>= 1) {
            for (u32 i = tid; i < NPAD; i += 1024) {
                u32 ixj = i ^ st;
                if (ixj > i) {
                    bool up = ((i & sz) == 0);
                    u64 a = skeys[i], b = skeys[ixj];
                    if (up ? (a < b) : (a > b)) { skeys[i] = b; skeys[ixj] = a; }
                }
            }
            __syncthreads();
        }
    }

    if (tid == 0) {                                   // build per-level lists
        int cnt[5] = {0, 0, 0, 0, 0};
        for (int r = 0; r < KSEL; ++r) {
            u64 key = skeys[r];
            int slot = (int)(key & 0x1FFFu);
            int l = slot < 1000 ? 0 : slot < 2000 ? 1 : slot < 3000 ? 2 : slot < 4000 ? 3 : 4;
            u32 valid = ((key >> 32) != 0ull) ? 1u : 0u;
            sortedSlot[(size_t)n * KSEL + r] = (u32)slot;
            int c = cnt[l]++;
            lvlList[((size_t)n * 5 + l) * 1000 + c] = (u32)slot | ((u32)r << 16) | (valid << 31);
        }
        for (int l = 0; l < 5; ++l) lvlCnt[n * 5 + l] = (u32)cnt[l];
    }
}

// ---------------------------------------------------------------------------
// 5) Per-(image,level) NMS: tiled 16x16 suppression mask (area-sum via WMMA,
//    intersection via VALU, wave32 ballot packing), wave-synchronous greedy.
// ---------------------------------------------------------------------------
__global__ __launch_bounds__(1024) void nms_kernel(const float* boxesG, const u32* lvlList,
                                                   const u32* lvlCnt, u32* keepOut) {
    int n = blockIdx.x, l = blockIdx.y;
    int K = (int)lvlCnt[n * 5 + l];
    int tid = threadIdx.x, lane = tid & 31, wave = tid >> 5;

    __shared__ float bx[1008], by[1008], bx2[1008], by2[1008];
    __shared__ float areaS[1008];
    __shared__ u16   mask16[256][64];
    __shared__ u16   keep16[64];
    __shared__ u16   rankArr[1008];
    __shared__ u16   validA[1024];

    for (int c = tid; c < 1008; c += 1024) {
        bx[c] = 0.f; by[c] = 0.f; bx2[c] = 0.f; by2[c] = 0.f;
        areaS[c] = 0.f; rankArr[c] = 0;
    }
    if (tid < 1024) validA[tid] = 0;
    __syncthreads();

    float off = (float)l * 1025.0f;                   // batched_nms level offset
    for (int c = tid; c < K; c += 1024) {
        u32 e = lvlList[((size_t)n * 5 + l) * 1000 + c];
        int slot = (int)(e & 0x1FFFu);
        int rank = (int)((e >> 16) & 0x1FFFu);
        int valid = (int)(e >> 31);
        const float* b = boxesG + ((size_t)n * KSEL + slot) * 4;
        float x1 = b[0] + off, y1 = b[1] + off, x2 = b[2] + off, y2 = b[3] + off;
        bx[c] = x1; by[c] = y1; bx2[c] = x2; by2[c] = y2;
        areaS[c] = (x2 - x1) * (y2 - y1);
        rankArr[c] = (u16)rank;
        validA[c] = (u16)valid;
    }
    __syncthreads();
    if (tid < 64) {
        u32 bits = 0;
        for (int b = 0; b < 16; ++b) bits |= ((u32)(validA[tid * 16 + b] & 1u)) << b;
        keep16[tid] = (u16)bits;
    }
    __syncthreads();

    int CT = (K + 15) >> 4;
    int W  = (K + 15) >> 4;
    for (int t = 0; t * 256 < K; ++t) {
        int r0 = t * 256;
        int R  = K - r0; if (R > 256) R = 256;
        int RT = (R + 15) >> 4;
        // --- mask tile compute: one 16x16 subtile per wave iteration ---
        for (int s = wave; s < RT * CT; s += 32) {
            int i16 = s / CT, j16 = s % CT;
            int rowBase = r0 + i16 * 16, colBase = j16 * 16;
            bool lo = lane < 16;
            int  lm = lane & 15;
            float asumTile[8];
#if defined(HAS_WMMA4)
            // A = [a_i, 1, 0, 0] (16x4), B = [1; a_j; 0; 0] (4x16)
            v2f Af, Bf;
            Af.x = lo ? areaS[rowBase + lm] : 0.f;    // K=0 / K=2
            Af.y = lo ? 1.0f : 0.f;                   // K=1 / K=3
            Bf.x = lo ? 1.0f : 0.f;
            Bf.y = lo ? areaS[colBase + lm] : 0.f;
            v8f acc = {};
            acc = __builtin_amdgcn_wmma_f32_16x16x4_f32(false, Af, false, Bf,
                                                        (short)0, acc, false, false);
            for (int k = 0; k < 8; ++k) asumTile[k] = acc[k];
#endif
            for (int k = 0; k < 8; ++k) {
                int r = lo ? k : (k + 8);
                int i = rowBase + r;
                int j = colBase + lm;
                bool inb = (i < K) && (j < K);
#if defined(HAS_WMMA4)
                float asum = asumTile[k];
#else
                float asum = areaS[i] + areaS[j];
#endif
                float xx1 = fmaxf(bx[i], bx[j]);
                float yy1 = fmaxf(by[i], by[j]);
                float xx2 = fminf(bx2[i], bx2[j]);
                float yy2 = fminf(by2[i], by2[j]);
                float iw = fmaxf(xx2 - xx1, 0.f), ih = fmaxf(yy2 - yy1, 0.f);
                float inter = iw * ih;
                float iou = inter / (asum - inter + 1e-9f);
                bool sup = inb && (iou > NMSTH);
                u32 b32 = (u32)__ballot((int)sup);    // wave32: 16 cols x 2 rows
                if (lane == 0) {
                    mask16[i16 * 16 + k][j16]     = (u16)(b32 & 0xFFFFu);
                    mask16[i16 * 16 + k + 8][j16] = (u16)(b32 >> 16);
                }
            }
        }
        __syncthreads();
        // --- wave-synchronous greedy over this row block ---
        if (tid < 32) {
            volatile u16* kp = keep16;
            for (int i = r0; i < r0 + R; ++i) {
                int wi = i >> 4;
                u32 kb = kp[wi];
                if ((kb >> (i & 15)) & 1u) {
                    for (int w = lane; w < W; w += 32) {
                        u32 allowed;
                        if (w < wi)       allowed = 0u;
                        else if (w == wi) allowed = (0xFFFFu << ((i & 15) + 1)) & 0xFFFFu;
                        else              allowed = 0xFFFFu;
                        u32 m = (u32)mask16[i - r0][w] & allowed;
                        if (m) kp[w] = (u16)((u32)kp[w] & ~m);
                    }
                }
            }
        }
        __syncthreads();
    }
    for (int c = tid; c < K; c += 1024) {
        u32 bit = ((u32)keep16[c >> 4] >> (c & 15)) & 1u;
        keepOut[(size_t)n * KSEL + rankArr[c]] = bit;
    }
}

// ---------------------------------------------------------------------------
// 6) Compact kept boxes in global score order into [POST,4], zero padded.
// ---------------------------------------------------------------------------
__global__ __launch_bounds__(256) void output_kernel(const float* boxesG, const u32* sortedSlot,
                                                     const u32* keepOut, float* out) {
    int n = blockIdx.x, tid = threadIdx.x;
    for (int i = tid; i < POST * 4; i += 256) out[(size_t)n * POST * 4 + i] = 0.f;
    __syncthreads();
    if (tid == 0) {
        int pos = 0;
        for (int r = 0; r < KSEL && pos < POST; ++r) {
            if (keepOut[(size_t)n * KSEL + r]) {
                int slot = (int)sortedSlot[(size_t)n * KSEL + r];
                const float* b = boxesG + ((size_t)n * KSEL + slot) * 4;
                float* o = out + ((size_t)n * POST + pos) * 4;
                o[0] = b[0]; o[1] = b[1]; o[2] = b[2]; o[3] = b[3];
                ++pos;
            }
        }
    }
}

// ---------------------------------------------------------------------------
extern "C" void kernel_launch(void* const* d_in, const int* in_sizes, int n_in,
                              void* d_out, int out_size, void* d_ws, size_t ws_size,
                              hipStream_t stream) {
    (void)in_sizes; (void)n_in; (void)out_size; (void)ws_size;
    const float* obj[5]; const float* del[5];
    for (int l = 0; l < 5; ++l) {               // setup_inputs dict order: obj_l, delta_l interleaved
        obj[l] = (const float*)d_in[2 * l];
        del[l] = (const float*)d_in[2 * l + 1];
    }
    const float* anchors = (const float*)d_in[10];
    float* out = (float*)d_out;

    char* w = (char*)d_ws;
    u32* thr       = (u32*)(w + 0);
    u32* neq       = (u32*)(w + 160);
    u32* gcnt      = (u32*)(w + 320);
    u32* cand      = (u32*)(w + 512);                    // 8*4768 u32
    u64* key64     = (u64*)(w + 153088);                 // 8*4768 u64
    float* boxesG  = (float*)(w + 458240);               // 8*4768*4 f32
    u32* sortedSlt = (u32*)(w + 1068544);                // 8*4768 u32
    u32* lvlList   = (u32*)(w + 1221120);                // 8*5*1000 u32
    u32* lvlCnt    = (u32*)(w + 1381120);                // 40 u32
    u32* keepOut   = (u32*)(w + 1381280);                // 8*4768 u32

    radix_kernel<<<dim3(8, 4), 256, 0, stream>>>(obj[0], obj[1], obj[2], obj[3], thr, neq, gcnt);
    collect_kernel<<<dim3(8, 5), 256, 0, stream>>>(obj[0], obj[1], obj[2], obj[3],
                                                   thr, neq, gcnt, cand);
    decode_kernel<<<(NIMG * KSEL + 255) / 256, 256, 0, stream>>>(
        obj[0], obj[1], obj[2], obj[3], obj[4],
        del[0], del[1], del[2], del[3], del[4],
        anchors, cand, boxesG, key64, keepOut);
    sort_kernel<<<8, 1024, 0, stream>>>(key64, sortedSlt, lvlList, lvlCnt);
    nms_kernel<<<dim3(8, 5), 1024, 0, stream>>>(boxesG, lvlList, lvlCnt, keepOut);
    output_kernel<<<8, 256, 0, stream>>>(boxesG, sortedSlt, keepOut, out);
}